// MultiScalePeriodicPatchEmbedding_23622320128504
// MI455X (gfx1250) — compile-verified
//
#include <hip/hip_runtime.h>
#include <math.h>

// ---------------- problem constants ----------------
#define LSEQ 512
#define NC   16
#define NB   8
#define DM   512
#define NF   256
#define NE   43
#define TP   1657
#define EMB_ELEMS 108593152   // 8*16*1657*512
#define LAM  0.01f

// workspace layout (float offsets)
#define WS_XS   0        // 4096
#define WS_XR   4096     // 2048
#define WS_XI   6144     // 2048
#define WS_O1R  8192     // 8192
#define WS_O1I  16384    // 8192
#define WS_AMP  24576    // 2048
#define WS_MASK 26624    // 344
#define WS_PEF  27008    // 4096
#define WS_PEP  31104    // 65536  -> total 96640 floats (~387 KB)

__constant__ int cP[NE]   = {2,3,4,5,6,7,8,9,10,11,12,13,14,15,16,17,18,19,20,21,22,23,24,25,26,
                             28,30,32,34,36,39,42,46,51,56,64,73,85,102,128,170,256,512};
__constant__ int cN[NE]   = {256,171,128,103,86,74,64,57,52,47,43,40,37,35,32,31,29,27,26,25,24,23,22,21,20,
                             19,18,16,16,15,14,13,12,11,10,8,8,7,6,4,4,2,1};
__constant__ int cOFF[NE] = {0,256,427,555,658,744,818,882,939,991,1038,1081,1121,1158,1193,1225,1256,1285,
                             1312,1338,1363,1387,1410,1432,1453,1473,1492,1510,1526,1542,1557,1571,1584,1596,
                             1607,1617,1625,1633,1640,1646,1650,1654,1656};

typedef float v2f __attribute__((ext_vector_type(2)));
typedef float v4f __attribute__((ext_vector_type(4)));
typedef float v8f __attribute__((ext_vector_type(8)));

// ---------- stage 1: xs = x @ w_start + b ----------
__global__ void k_start(const float* __restrict__ x, const float* __restrict__ wst,
                        const float* __restrict__ bst, float* __restrict__ ws) {
  int id = blockIdx.x * 256 + threadIdx.x;            // [0, 4096)
  const float* xe = x + (size_t)id * NC;
  float s = 0.f;
#pragma unroll
  for (int c = 0; c < NC; ++c) s += xe[c] * wst[c];
  ws[WS_XS + id] = s + bst[0];
}

// ---------- stage 2: direct DFT (ortho), bins 1..256 ----------
__global__ void k_dft(float* __restrict__ ws) {
  int tid = threadIdx.x, lane = tid & 31, wv = tid >> 5;
  int bin = blockIdx.x * 8 + wv;                      // [0, 2048)
  int b = bin >> 8, k = (bin & 255) + 1;
  const float* xs = ws + WS_XS + b * LSEQ;
  float sr = 0.f, si = 0.f;
  for (int t = lane; t < LSEQ; t += 32) {
    int ph = (k * t) & (LSEQ - 1);
    float ang = (float)ph * (6.283185307179586f / (float)LSEQ);
    float sn, cs; __sincosf(ang, &sn, &cs);
    float v = xs[t];
    sr += v * cs; si -= v * sn;
  }
  for (int off = 16; off; off >>= 1) { sr += __shfl_down(sr, off); si += __shfl_down(si, off); }
  if (lane == 0) {
    const float nrm = 0.04419417382415922f;           // 1/sqrt(512)
    ws[WS_XR + bin] = sr * nrm;
    ws[WS_XI + bin] = si * nrm;
  }
}

// ---------- stage 3: complex layer 1 + relu ----------
__global__ void k_mlp1(const float* __restrict__ w1, const float* __restrict__ b1,
                       float* __restrict__ ws) {
  int id = blockIdx.x * 256 + threadIdx.x;            // [0, 8192)
  int b = id >> 10, j = id & 1023;
  const float* xr = ws + WS_XR + b * NF;
  const float* xi = ws + WS_XI + b * NF;
  const float* wr = w1 + j;                           // w1[0][k][j], stride 1024
  const float* wi = w1 + NF * 1024 + j;               // w1[1][k][j]
  float sr = 0.f, si = 0.f;
  for (int k = 0; k < NF; ++k) {
    float ar = xr[k], ai = xi[k];
    float wrv = wr[k * 1024], wiv = wi[k * 1024];
    sr += ar * wrv - ai * wiv;
    si += ai * wrv + ar * wiv;
  }
  ws[WS_O1R + id] = fmaxf(sr + b1[j], 0.f);
  ws[WS_O1I + id] = fmaxf(si + b1[1024 + j], 0.f);
}

__device__ __forceinline__ float softshrink(float v) {
  return (v > LAM) ? (v - LAM) : ((v < -LAM) ? (v + LAM) : 0.f);
}

// ---------- stage 4: complex layer 2 + softshrink + amplitude ----------
__global__ void k_mlp2(const float* __restrict__ w2, const float* __restrict__ b2,
                       float* __restrict__ ws) {
  int id = blockIdx.x * 256 + threadIdx.x;            // [0, 2048)
  int b = id >> 8, j = id & 255;
  const float* ar = ws + WS_O1R + b * 1024;
  const float* ai = ws + WS_O1I + b * 1024;
  float sr = 0.f, si = 0.f;
  for (int k = 0; k < 1024; ++k) {
    float wrv = w2[k * NF + j];
    float wiv = w2[1024 * NF + k * NF + j];
    float vr = ar[k], vi = ai[k];
    sr += vr * wrv - vi * wiv;
    si += vi * wrv + vr * wiv;
  }
  sr = softshrink(sr + b2[j]);
  si = softshrink(si + b2[NF + j]);
  ws[WS_AMP + id] = sqrtf(sr * sr + si * si + 1e-12f);
}

// ---------- stage 5: logits, top-5 softmax -> gates + mask ----------
__global__ void k_gate(const float* __restrict__ wg, float* __restrict__ out,
                       float* __restrict__ ws) {
  __shared__ float lg[NE];
  __shared__ float gv[5];
  __shared__ int   gi[5];
  int b = blockIdx.x, e = threadIdx.x;
  const float* amp = ws + WS_AMP + b * NF;
  if (e < NE) {
    float s = 0.f;
    for (int k = 0; k < NF; ++k) s += amp[k] * wg[k * NE + e];
    lg[e] = s;
  }
  __syncthreads();
  if (e == 0) {
    bool used[NE] = {false};
    for (int t = 0; t < 5; ++t) {
      int bi = 0; float bv = -3.4e38f;
      for (int q = 0; q < NE; ++q)
        if (!used[q] && lg[q] > bv) { bv = lg[q]; bi = q; }
      used[bi] = true; gv[t] = bv; gi[t] = bi;
    }
    float mx = gv[0], sum = 0.f, ex[5];
    for (int t = 0; t < 5; ++t) { ex[t] = __expf(gv[t] - mx); sum += ex[t]; }
    for (int t = 0; t < 5; ++t) gv[t] = ex[t] / sum;
  }
  __syncthreads();
  if (e < NE) {
    float g = 0.f;
    for (int t = 0; t < 5; ++t) if (gi[t] == e) g = gv[t];
    out[(size_t)EMB_ELEMS + b * NE + e] = g;
    ws[WS_MASK + b * NE + e] = (g > 0.f) ? 1.f : 0.f;
  }
}

// ---------- stage 6: positional-embedding tables ----------
__global__ void k_pe(float* __restrict__ ws) {
  int id = blockIdx.x * 256 + threadIdx.x;            // [0, 69632)
  if (id < 4096) {                                    // pe_feat [16,256]
    int c = id >> 8, d = id & 255;
    float inv = __expf(-logf(10000.f) * (float)(2 * (d >> 1)) / 256.f);
    float ang = (float)c * inv;
    ws[WS_PEF + id] = (d & 1) ? cosf(ang) : sinf(ang);
  } else if (id < 4096 + 65536) {                     // pe_pos [256,256]
    int id2 = id - 4096;
    int j = id2 >> 8, d = id2 & 255;
    float inv = __expf(-logf(10000.f) * (float)(2 * (d >> 1)) / 256.f);
    float ang = (float)j * inv;
    ws[WS_PEP + id2] = (d & 1) ? cosf(ang) : sinf(ang);
  }
}

// ---------- stage 7: per-expert patch embedding via V_WMMA_F32_16X16X4_F32 ----------
// grid (256, 43, 8) x 256 threads. Block = (rowTile rt, expert i, batch b).
// M = C*n rows flattened r = c*n + j (always multiple of 16). K = p.
// Each wave owns 64 consecutive d (4 accumulators); A fragment shared across the 4 tiles.
__global__ void k_emb(const float* __restrict__ x, const float* __restrict__ Wv,
                      const float* __restrict__ ws, float* __restrict__ out) {
  const int i = blockIdx.y, b = blockIdx.z, rt = blockIdx.x;
  const int p = cP[i], n = cN[i], J0 = cOFF[i];
  if (rt >= n) return;
  const int tid = threadIdx.x;

  if (ws[WS_MASK + b * NE + i] == 0.f) {              // inactive expert: stream zeros (NT)
    int row = tid >> 4, q = tid & 15;
    int lrow = rt * 16 + row;
    int cc = lrow / n, jj = lrow - cc * n;
    v4f* dst = (v4f*)(out + (((size_t)(b * NC + cc) * TP + (J0 + jj)) << 9));
    v4f z = {0.f, 0.f, 0.f, 0.f};
#pragma unroll
    for (int w = 0; w < 8; ++w) __builtin_nontemporal_store(z, &dst[w * 16 + q]);
    return;
  }

  const int lane = tid & 31, wv = tid >> 5;
  const int m = lane & 15;
  const int lrow = rt * 16 + m;
  const int c0 = lrow / n;
  const int j0 = lrow - c0 * n;
  const int jp = j0 * p;
  const int koff = (lane >> 4) << 1;                  // 0 or 2 (A/B lane-half K offset)
  const float* xb  = x  + (size_t)b * (LSEQ * NC);
  const float* wvE = Wv + (size_t)i * (512 * 512);
  const float* peF = ws + WS_PEF;
  const float* peP = ws + WS_PEP;
  const int bBase = koff * DM + wv * 64 + m;          // B fragment lane offset

  int ccr[8], jjr[8];                                 // row -> (c, j) for the store phase
#pragma unroll
  for (int r = 0; r < 8; ++r) {
    int row = rt * 16 + r + ((lane >= 16) ? 8 : 0);
    ccr[r] = row / n; jjr[r] = row - ccr[r] * n;
  }

  v8f acc[4];
#pragma unroll
  for (int dti = 0; dti < 4; ++dti) acc[dti] = (v8f){0.f,0.f,0.f,0.f,0.f,0.f,0.f,0.f};

  // ---- main K loop: full chunks, branch-free ----
  const int pfull = p & ~3;
  for (int kk = 0; kk < pfull; kk += 4) {
    int t0 = jp + kk + koff; if (t0 > LSEQ - 1) t0 = LSEQ - 1;   // edge padding
    int t1 = t0 + 1;        if (t1 > LSEQ - 1) t1 = LSEQ - 1;
    v2f av = { xb[t0 * NC + c0], xb[t1 * NC + c0] };
    const float* wk = wvE + kk * DM + bBase;
#pragma unroll
    for (int dti = 0; dti < 4; ++dti) {
      v2f bv = { wk[dti * 16], wk[dti * 16 + DM] };
      acc[dti] = __builtin_amdgcn_wmma_f32_16x16x4_f32(false, av, false, bv,
                                                       (short)0, acc[dti], false, false);
    }
  }

  // ---- tail chunk (p % 4): unconditional loads + register selects (no EXEC branching) ----
  if (p & 3) {
    int k0 = pfull + koff, k1 = k0 + 1;
    int t0 = jp + k0; if (t0 > LSEQ - 1) t0 = LSEQ - 1;
    int t1 = jp + k1; if (t1 > LSEQ - 1) t1 = LSEQ - 1;
    float a0 = xb[t0 * NC + c0];
    float a1 = xb[t1 * NC + c0];
    a0 = (k0 < p) ? a0 : 0.f;                         // zero A past K=p
    a1 = (k1 < p) ? a1 : 0.f;
    v2f av = { a0, a1 };
    const float* wk = wvE + pfull * DM + bBase;       // rows >= p are finite randn * 0
#pragma unroll
    for (int dti = 0; dti < 4; ++dti) {
      v2f bv = { wk[dti * 16], wk[dti * 16 + DM] };
      acc[dti] = __builtin_amdgcn_wmma_f32_16x16x4_f32(false, av, false, bv,
                                                       (short)0, acc[dti], false, false);
    }
  }

  // ---- store: +pos-emb, non-temporal (output is write-once, larger than L2) ----
#pragma unroll
  for (int r = 0; r < 8; ++r) {
    size_t rowBase = ((size_t)(b * NC + ccr[r]) * TP + (J0 + jjr[r])) * (size_t)DM;
#pragma unroll
    for (int dti = 0; dti < 4; ++dti) {
      int d = wv * 64 + dti * 16 + m;
      float pe = (d < 256) ? peF[ccr[r] * 256 + d] : peP[jjr[r] * 256 + (d - 256)];
      __builtin_nontemporal_store(acc[dti][r] + pe, &out[rowBase + d]);
    }
  }
}

extern "C" void kernel_launch(void* const* d_in, const int* in_sizes, int n_in,
                              void* d_out, int out_size, void* d_ws, size_t ws_size,
                              hipStream_t stream) {
  const float* x      = (const float*)d_in[0];
  const float* wstart = (const float*)d_in[1];
  const float* bstart = (const float*)d_in[2];
  const float* w1     = (const float*)d_in[3];
  const float* b1     = (const float*)d_in[4];
  const float* w2     = (const float*)d_in[5];
  const float* b2     = (const float*)d_in[6];
  const float* wgate  = (const float*)d_in[7];
  const float* Wv     = (const float*)d_in[8];
  float* out = (float*)d_out;
  float* ws  = (float*)d_ws;

  k_start<<<16, 256, 0, stream>>>(x, wstart, bstart, ws);
  k_dft  <<<256, 256, 0, stream>>>(ws);
  k_mlp1 <<<32, 256, 0, stream>>>(w1, b1, ws);
  k_mlp2 <<<8, 256, 0, stream>>>(w2, b2, ws);
  k_gate <<<8, 64, 0, stream>>>(wgate, out, ws);
  k_pe   <<<272, 256, 0, stream>>>(ws);
  k_emb  <<<dim3(256, NE, NB), 256, 0, stream>>>(x, Wv, ws, out);
}